// DQN_48129403519175
// MI455X (gfx1250) — compile-verified
//
#include <hip/hip_runtime.h>
#include <math.h>

#define N_NODES 30000
#define N_EDGES 300000
#define E_TOT   (N_EDGES + N_NODES)
#define ACTION_K 3000
#define NHEAD 4
#define BN_EPS 1e-5f

typedef __attribute__((ext_vector_type(2))) float v2f;
typedef __attribute__((ext_vector_type(8))) float v8f;

// ---------------- float atomic max (sign-split int trick) ----------------
__device__ __forceinline__ void atomicMaxF(float* addr, float v) {
  if (v >= 0.0f) atomicMax((int*)addr, __float_as_int(v));
  else           atomicMin((unsigned int*)addr, __float_as_uint(v));
}

// ---------------- utility fills ----------------
__global__ void fill_f32(float* __restrict__ p, float v, int n) {
  int t = blockIdx.x * blockDim.x + threadIdx.x;
  if (t < n) p[t] = v;
}
__global__ void fill_i32(int* __restrict__ p, int v, int n) {
  int t = blockIdx.x * blockDim.x + threadIdx.x;
  if (t < n) p[t] = v;
}
__global__ void fill_bias(float* __restrict__ p, const float* __restrict__ bias, int total, int C) {
  int t = blockIdx.x * blockDim.x + threadIdx.x;
  if (t < total) p[t] = bias[t % C];
}

// ---------------- edge list with self loops ----------------
__global__ void build_edges(const int* __restrict__ ei, int* __restrict__ src, int* __restrict__ dst) {
  int t = blockIdx.x * blockDim.x + threadIdx.x;
  if (t >= E_TOT) return;
  if (t < N_EDGES) { src[t] = ei[t]; dst[t] = ei[N_EDGES + t]; }
  else             { int v = t - N_EDGES; src[t] = v; dst[t] = v; }
}

// ---------------- f32 WMMA GEMM: O[M x N] = X[M x K](row stride ldx) @ W[K x N] ----------------
// one wave computes a 16x16 tile with V_WMMA_F32_16X16X4_F32, K stepped by 4; scalar K remainder.
__global__ void gemm_wmma_f32(const float* __restrict__ X, int ldx,
                              const float* __restrict__ W,
                              float* __restrict__ O, int M, int N, int K) {
  int wave = (blockIdx.x * blockDim.x + threadIdx.x) >> 5;
  int lane = threadIdx.x & 31;
  int tilesN = N >> 4;
  int tilesM = (M + 15) >> 4;
  if (wave >= tilesM * tilesN) return;           // wave-uniform exit (EXEC stays all-ones for WMMA)
  int tm = wave / tilesN, tn = wave % tilesN;
  int row0 = tm << 4, col0 = tn << 4;
  int hi2 = lane >> 4;                           // 0: K pair {0,1}, 1: K pair {2,3}
  int l16 = lane & 15;

  v8f c = {0.f,0.f,0.f,0.f,0.f,0.f,0.f,0.f};
  int k = 0;
  for (; k + 4 <= K; k += 4) {
    v2f a, b;
    const float* xr = X + (size_t)(row0 + l16) * ldx + k + hi2 * 2;
    a.x = xr[0]; a.y = xr[1];
    const float* wr = W + (size_t)(k + hi2 * 2) * N + col0 + l16;
    b.x = wr[0]; b.y = wr[N];
    c = __builtin_amdgcn_wmma_f32_16x16x4_f32(false, a, false, b, (short)0, c, false, false);
  }
  for (; k < K; ++k) {                            // rank-1 remainder (layer 1: K=5)
    float wv = W[(size_t)k * N + col0 + l16];
#pragma unroll
    for (int r = 0; r < 8; ++r) {
      int row = row0 + r + hi2 * 8;
      c[r] += X[(size_t)row * ldx + k] * wv;
    }
  }
#pragma unroll
  for (int r = 0; r < 8; ++r) {
    int row = row0 + r + hi2 * 8;
    O[(size_t)row * N + col0 + l16] = c[r];
  }
}

// ---------------- per-(node,head) attention scores ----------------
__global__ void scores_k(const float* __restrict__ hbuf, const float* __restrict__ as_,
                         const float* __restrict__ ad_, float* __restrict__ esrc,
                         float* __restrict__ edst, int n, int H, int C) {
  int t = blockIdx.x * blockDim.x + threadIdx.x;
  if (t >= n * H) return;
  int h = t % H;
  const float* hp = hbuf + (size_t)t * C;        // t == node*H + h, layout (n,H,C)
  float a = 0.f, b = 0.f;
  for (int cc = 0; cc < C; ++cc) { float v = hp[cc]; a += v * as_[h * C + cc]; b += v * ad_[h * C + cc]; }
  esrc[t] = a; edst[t] = b;
}

// ---------------- edge passes ----------------
__global__ void edge_max(const int* __restrict__ src, const int* __restrict__ dst,
                         const float* __restrict__ esrc, const float* __restrict__ edst,
                         float* __restrict__ ebuf, float* __restrict__ mbuf, int E, int H) {
  int t = blockIdx.x * blockDim.x + threadIdx.x;
  if (t >= E * H) return;
  int h = t & (NHEAD - 1); int e = t >> 2;
  int s = src[e], d = dst[e];
  float v = esrc[s * H + h] + edst[d * H + h];
  v = v > 0.f ? v : 0.2f * v;                    // leaky_relu(0.2)
  ebuf[t] = v;
  atomicMaxF(&mbuf[d * H + h], v);
}

__global__ void edge_exp(const int* __restrict__ dst, float* __restrict__ ebuf,
                         const float* __restrict__ mbuf, float* __restrict__ sbuf, int E, int H) {
  int t = blockIdx.x * blockDim.x + threadIdx.x;
  if (t >= E * H) return;
  int h = t & (NHEAD - 1); int e = t >> 2;
  int d = dst[e];
  float ex = expf(ebuf[t] - mbuf[d * H + h]);
  ebuf[t] = ex;
  atomicAdd(&sbuf[d * H + h], ex);
}

// per-(edge,channel) thread; sums all heads (with 1/H) into one atomic per element
__global__ void edge_agg(const int* __restrict__ src, const int* __restrict__ dst,
                         const float* __restrict__ hbuf, const float* __restrict__ ebuf,
                         const float* __restrict__ sbuf, float* __restrict__ nodebuf,
                         int E, int H, int C) {
  int t = blockIdx.x * blockDim.x + threadIdx.x;
  if (t >= E * C) return;
  int c = t % C; int e = t / C;
  int s = src[e], d = dst[e];
  float acc = 0.f;
  for (int h = 0; h < NHEAD; ++h) {
    float alpha = ebuf[e * H + h] / (sbuf[d * H + h] + 1e-16f);
    acc += alpha * hbuf[(size_t)s * H * C + h * C + c];
  }
  atomicAdd(&nodebuf[(size_t)d * C + c], acc * (1.0f / NHEAD));
}

// ---------------- batch norm stats / apply + ELU + dropout ----------------
__global__ void bn_stats(const float* __restrict__ x, int n, int C,
                         float* __restrict__ mu, float* __restrict__ inv) {
  int c = blockIdx.x;
  __shared__ float s1[256], s2[256];
  float a = 0.f, b = 0.f;
  for (int i = threadIdx.x; i < n; i += 256) { float v = x[(size_t)i * C + c]; a += v; b += v * v; }
  s1[threadIdx.x] = a; s2[threadIdx.x] = b; __syncthreads();
  for (int s = 128; s > 0; s >>= 1) {
    if (threadIdx.x < s) { s1[threadIdx.x] += s1[threadIdx.x + s]; s2[threadIdx.x] += s2[threadIdx.x + s]; }
    __syncthreads();
  }
  if (threadIdx.x == 0) {
    float m = s1[0] / n;
    float var = s2[0] / n - m * m;
    mu[c] = m; inv[c] = rsqrtf(var + BN_EPS);
  }
}

__global__ void bn_apply(const float* __restrict__ x, const float* __restrict__ mu,
                         const float* __restrict__ inv, const float* __restrict__ g,
                         const float* __restrict__ be, const float* __restrict__ u,
                         float* __restrict__ out, int total, int C, int do_elu) {
  int t = blockIdx.x * blockDim.x + threadIdx.x;
  if (t >= total) return;
  int c = t % C;
  float v = (x[t] - mu[c]) * inv[c] * g[c] + be[c];
  if (do_elu) v = v > 0.f ? v : expm1f(v);
  v *= (u[t] > 0.5f) ? 2.0f : 0.0f;
  out[t] = v;
}

// ---------------- column mean (pooled) ----------------
__global__ void col_mean(const float* __restrict__ x, int n, int C, float* __restrict__ out) {
  int c = blockIdx.x;
  __shared__ float s1[256];
  float a = 0.f;
  for (int i = threadIdx.x; i < n; i += 256) a += x[(size_t)i * C + c];
  s1[threadIdx.x] = a; __syncthreads();
  for (int s = 128; s > 0; s >>= 1) { if (threadIdx.x < s) s1[threadIdx.x] += s1[threadIdx.x + s]; __syncthreads(); }
  if (threadIdx.x == 0) out[c] = s1[0] / n;
}

// ---------------- ordered compaction of flag==1.0 (wave32 ballot) ----------------
__global__ void compact_idx(const float* __restrict__ x, int n, int ld, int col,
                            int* __restrict__ idx, int maxk) {
  int lane = threadIdx.x;     // 32 threads, 1 block
  int base = 0;
  for (int start = 0; start < n; start += 32) {
    int i = start + lane;
    bool pred = (i < n) && (x[(size_t)i * ld + col] == 1.0f);
    unsigned long long m = __ballot((int)pred);
    unsigned m32 = (unsigned)m;
    int prefix = __popc(m32 & ((1u << lane) - 1u));
    if (pred && base + prefix < maxk) idx[base + prefix] = i;
    base += __popc(m32);
    if (base >= maxk) break;   // base is lane-uniform
  }
}

// ---------------- collapse MLP head: K1 = A1@A2@A3, K2 = (ab1@A2 + ab2)@A3 + ab3 ----------------
__global__ void head_consts(const float* __restrict__ A1, const float* __restrict__ ab1,
                            const float* __restrict__ A2, const float* __restrict__ ab2,
                            const float* __restrict__ A3, const float* __restrict__ ab3,
                            float* __restrict__ k12) {
  int j = threadIdx.x;  // 64 threads
  float t = 0.f, u = 0.f;
  for (int c = 0; c < 128; ++c) { float a2 = A2[c * 64 + j]; t += A1[c] * a2; u += ab1[c] * a2; }
  __shared__ float sT[64], sU[64];
  sT[j] = t * A3[j];
  sU[j] = (u + ab2[j]) * A3[j];
  __syncthreads();
  for (int s = 32; s > 0; s >>= 1) { if (j < s) { sT[j] += sT[j + s]; sU[j] += sU[j + s]; } __syncthreads(); }
  if (j == 0) { k12[0] = sT[0]; k12[1] = sU[0] + ab3[0]; }
}

// ---------------- logits: tanh(dot(h[idx], pooled)*K1 + K2) ----------------
__global__ void logits_k(const float* __restrict__ h, const int* __restrict__ idx,
                         const float* __restrict__ pooled, const float* __restrict__ k12,
                         float* __restrict__ out) {
  int t = blockIdx.x * blockDim.x + threadIdx.x;
  if (t >= ACTION_K) return;
  int node = idx[t];
  float a = 0.f;
  for (int c = 0; c < 32; ++c) a += h[(size_t)node * 32 + c] * pooled[c];
  out[t] = tanhf(a * k12[0] + k12[1]);
}

// ---------------- softmax over 3000 logits -> out[3000..5999] ----------------
__global__ void softmax_k(float* __restrict__ out) {
  __shared__ float red[1024];
  int t = threadIdx.x;
  float mx = -INFINITY;
  for (int i = t; i < ACTION_K; i += 1024) mx = fmaxf(mx, out[i]);
  red[t] = mx; __syncthreads();
  for (int s = 512; s > 0; s >>= 1) { if (t < s) red[t] = fmaxf(red[t], red[t + s]); __syncthreads(); }
  float m = red[0]; __syncthreads();
  float sum = 0.f;
  for (int i = t; i < ACTION_K; i += 1024) sum += expf(out[i] - m);
  red[t] = sum; __syncthreads();
  for (int s = 512; s > 0; s >>= 1) { if (t < s) red[t] += red[t + s]; __syncthreads(); }
  float inv = 1.0f / red[0];
  for (int i = t; i < ACTION_K; i += 1024) out[ACTION_K + i] = expf(out[i] - m) * inv;
}

// =====================================================================
extern "C" void kernel_launch(void* const* d_in, const int* in_sizes, int n_in,
                              void* d_out, int out_size, void* d_ws, size_t ws_size,
                              hipStream_t stream) {
  const float* x   = (const float*)d_in[0];
  const int*   ei  = (const int*)  d_in[1];
  const float* u1  = (const float*)d_in[2];
  const float* u2  = (const float*)d_in[3];
  const float* u3  = (const float*)d_in[4];
  const float* W1  = (const float*)d_in[5];
  const float* as1 = (const float*)d_in[6];
  const float* ad1 = (const float*)d_in[7];
  const float* b1  = (const float*)d_in[8];
  const float* g1  = (const float*)d_in[9];
  const float* be1 = (const float*)d_in[10];
  const float* W2  = (const float*)d_in[11];
  const float* as2 = (const float*)d_in[12];
  const float* ad2 = (const float*)d_in[13];
  const float* b2  = (const float*)d_in[14];
  const float* g2  = (const float*)d_in[15];
  const float* be2 = (const float*)d_in[16];
  const float* W3  = (const float*)d_in[17];
  const float* as3 = (const float*)d_in[18];
  const float* ad3 = (const float*)d_in[19];
  const float* b3  = (const float*)d_in[20];
  const float* g3  = (const float*)d_in[21];
  const float* be3 = (const float*)d_in[22];
  const float* A1  = (const float*)d_in[23];
  const float* ab1 = (const float*)d_in[24];
  const float* A2  = (const float*)d_in[25];
  const float* ab2 = (const float*)d_in[26];
  const float* A3  = (const float*)d_in[27];
  const float* ab3 = (const float*)d_in[28];
  float* out = (float*)d_out;

  // ---- carve workspace ----
  char* ws = (char*)d_ws;
  size_t off = 0;
  auto carve = [&](size_t bytes) -> void* {
    void* p = ws + off;
    off = (off + bytes + 255) & ~(size_t)255;
    return p;
  };
  int*   eiS     = (int*)  carve((size_t)E_TOT * 4);
  int*   eiD     = (int*)  carve((size_t)E_TOT * 4);
  float* hbuf    = (float*)carve((size_t)N_NODES * 512 * 4);   // GEMM out (n,H,C)
  float* nodebuf = (float*)carve((size_t)N_NODES * 128 * 4);   // aggregated + bias
  float* nf      = (float*)carve((size_t)N_NODES * 128 * 4);   // post BN/act/drop
  float* esrc    = (float*)carve((size_t)N_NODES * NHEAD * 4);
  float* edstS   = (float*)carve((size_t)N_NODES * NHEAD * 4);
  float* mbuf    = (float*)carve((size_t)N_NODES * NHEAD * 4);
  float* sbuf    = (float*)carve((size_t)N_NODES * NHEAD * 4);
  float* ebuf    = (float*)carve((size_t)E_TOT * NHEAD * 4);
  float* mu      = (float*)carve(512 * 4);
  float* inv     = (float*)carve(512 * 4);
  float* pooled  = (float*)carve(32 * 4);
  float* k12     = (float*)carve(2 * 4);
  int*   idx     = (int*)  carve(ACTION_K * 4);

  const int T = 256;
  auto blocks = [](long long n, int t) { return (unsigned)((n + t - 1) / t); };

  build_edges<<<blocks(E_TOT, T), T, 0, stream>>>(ei, eiS, eiD);

  auto gat_layer = [&](const float* X, int ldx, int K,
                       const float* W, const float* aS, const float* aD, const float* bias,
                       const float* gamma, const float* beta, const float* u,
                       int C, int doElu) {
    const int H = NHEAD, N = H * C;
    long long tiles = (long long)(N_NODES / 16) * (N / 16);
    gemm_wmma_f32<<<blocks(tiles * 32, T), T, 0, stream>>>(X, ldx, W, hbuf, N_NODES, N, K);
    scores_k<<<blocks((long long)N_NODES * H, T), T, 0, stream>>>(hbuf, aS, aD, esrc, edstS, N_NODES, H, C);
    fill_f32<<<blocks((long long)N_NODES * H, T), T, 0, stream>>>(mbuf, -INFINITY, N_NODES * H);
    fill_f32<<<blocks((long long)N_NODES * H, T), T, 0, stream>>>(sbuf, 0.0f, N_NODES * H);
    fill_bias<<<blocks((long long)N_NODES * C, T), T, 0, stream>>>(nodebuf, bias, N_NODES * C, C);
    edge_max<<<blocks((long long)E_TOT * H, T), T, 0, stream>>>(eiS, eiD, esrc, edstS, ebuf, mbuf, E_TOT, H);
    edge_exp<<<blocks((long long)E_TOT * H, T), T, 0, stream>>>(eiD, ebuf, mbuf, sbuf, E_TOT, H);
    edge_agg<<<blocks((long long)E_TOT * C, T), T, 0, stream>>>(eiS, eiD, hbuf, ebuf, sbuf, nodebuf, E_TOT, H, C);
    bn_stats<<<C, T, 0, stream>>>(nodebuf, N_NODES, C, mu, inv);
    bn_apply<<<blocks((long long)N_NODES * C, T), T, 0, stream>>>(nodebuf, mu, inv, gamma, beta, u, nf,
                                                                  N_NODES * C, C, doElu);
  };

  // layer 1: in_x = x[:, :5] (row stride 6), out C=128, ELU
  gat_layer(x, 6, 5,   W1, as1, ad1, b1, g1, be1, u1, 128, 1);
  // layer 2: nf(30000x128) -> C=64, ELU   (nf read fully by GEMM before rewrite: safe in-place ping)
  gat_layer(nf, 128, 128, W2, as2, ad2, b2, g2, be2, u2, 64, 1);
  // layer 3: nf(30000x64) -> C=32, no ELU
  gat_layer(nf, 64, 64,  W3, as3, ad3, b3, g3, be3, u3, 32, 0);

  // pooled = column mean of h (30000 x 32)
  col_mean<<<32, T, 0, stream>>>(nf, N_NODES, 32, pooled);
  // ordered nonzero(x[:,3]==1.0, size=3000, fill=0)
  fill_i32<<<blocks(ACTION_K, T), T, 0, stream>>>(idx, 0, ACTION_K);
  compact_idx<<<1, 32, 0, stream>>>(x, N_NODES, 6, 3, idx, ACTION_K);
  // collapse affine head to tanh(a*K1 + K2)
  head_consts<<<1, 64, 0, stream>>>(A1, ab1, A2, ab2, A3, ab3, k12);
  // logits then softmax
  logits_k<<<blocks(ACTION_K, T), T, 0, stream>>>(nf, idx, pooled, k12, out);
  softmax_k<<<1, 1024, 0, stream>>>(out);
}